// DA_Adapter_text_79559974191379
// MI455X (gfx1250) — compile-verified
//
#include <hip/hip_runtime.h>

// ---------------------------------------------------------------------------
// MI455X (gfx1250) adapter kernel: LN -> 768x16 -> ReLU -> 16x768 -> scale+res
// Memory-bound (~201MB @ 23.3TB/s => ~8.6us floor). fp32 WMMA 16x16x4 keeps
// full precision while letting the matrix pipes do the skinny GEMMs.
// One wave32 owns one 16-token tile end-to-end; LayerNorm is folded into the
// down-projection so WMMA A-frags read raw x directly from global memory.
// ---------------------------------------------------------------------------

typedef float v2f __attribute__((ext_vector_type(2)));
typedef float v4f __attribute__((ext_vector_type(4)));
typedef float v8f __attribute__((ext_vector_type(8)));

#define D_MODEL 768
#define BN      16
#define SCALE_F 0.1f
#define LN_EPS  1e-5f

// LDS layout (float offsets), total 27680 floats = 110720 bytes (< 320KB WGP LDS)
#define OFF_WG   0        // 768x16 gamma-scaled W_down, WMMA-B swizzled (12288)
#define OFF_WU   12288    // 16x768 W_up, WMMA-B swizzled per 16-col tile (12288)
#define OFF_BUP  24576    // b_up (768)
#define OFF_C1   25344    // gamma^T W_down (16)
#define OFF_C2   25360    // beta^T W_down + b_down (16)
#define OFF_DOWN 25376    // per-wave 16x16 down matrix (8 waves * 256)
#define OFF_MU   27424    // per-wave mean[16] (8 waves * 16)
#define OFF_RS   27552    // per-wave rstd[16] (8 waves * 16)
#define LDS_FLOATS 27680

__global__ __launch_bounds__(256)
void adapter_wmma_kernel(const float* __restrict__ x,
                         const float* __restrict__ gamma,
                         const float* __restrict__ beta,
                         const float* __restrict__ wdown,   // [768][16]
                         const float* __restrict__ bdown,   // [16]
                         const float* __restrict__ wup,     // [16][768]
                         const float* __restrict__ bup,     // [768]
                         float* __restrict__ out,
                         int nTokens)
{
    extern __shared__ float smem[];
    const int tid     = threadIdx.x;
    const int lane    = tid & 31;
    const int wave    = tid >> 5;
    const int halfSel = lane >> 4;   // 0: lanes 0-15, 1: lanes 16-31
    const int laneN   = lane & 15;

    // ---- Stage weights into LDS, pre-swizzled for the WMMA B-matrix layout.
    // B(4x16) per op: lanes 0-15 hold (K=k0,k0+1), lanes 16-31 hold (K=k0+2,k0+3),
    // N = lane&15. We store each lane's (K,K+1) pair contiguously -> ds_load_b64.
    for (int i = tid; i < D_MODEL * BN; i += 256) {
        int d = i >> 4, n = i & 15;                 // wdown[d][n]
        float wg = gamma[d] * wdown[i];             // fold LN gamma into W_down
        int r = d & 3, kb = d >> 2;
        smem[OFF_WG + kb * 64 + ((r >> 1) * 16 + n) * 2 + (r & 1)] = wg;
    }
    for (int i = tid; i < BN * D_MODEL; i += 256) {
        int k = i / D_MODEL, n = i - k * D_MODEL;   // wup[k][n]
        int nt = n >> 4, nn = n & 15;
        int r = k & 3, kb = k >> 2;
        smem[OFF_WU + (nt * 4 + kb) * 64 + ((r >> 1) * 16 + nn) * 2 + (r & 1)] = wup[i];
    }
    for (int i = tid; i < D_MODEL; i += 256) smem[OFF_BUP + i] = bup[i];
    if (tid < BN) {                                  // LN-fold correction vectors
        float c1 = 0.f, c2 = 0.f;
        for (int d = 0; d < D_MODEL; ++d) {
            float w = wdown[d * BN + tid];
            c1 += gamma[d] * w;
            c2 += beta[d] * w;
        }
        smem[OFF_C1 + tid] = c1;
        smem[OFF_C2 + tid] = c2 + bdown[tid];
    }
    __syncthreads();

    float* ldsDown = smem + OFF_DOWN + wave * 256;
    float* ldsMu   = smem + OFF_MU   + wave * 16;
    float* ldsRs   = smem + OFF_RS   + wave * 16;

    const float c1k = smem[OFF_C1 + laneN];
    const float c2k = smem[OFF_C2 + laneN];

    const int nTiles     = nTokens >> 4;            // 16 tokens per tile
    const int globalWave = (int)blockIdx.x * 8 + wave;
    const int totalWaves = (int)gridDim.x * 8;

    for (int tile = globalWave; tile < nTiles; tile += totalWaves) {
        const float* xt = x + (size_t)tile * (16 * D_MODEL);

        // Prefetch the next tile this wave will touch (48KB = 768 lines;
        // 24 global_prefetch_b8 x 32 lanes x 64B covers it).
        int nxt = tile + totalWaves;
        if (nxt < nTiles) {
            const char* p = (const char*)(x + (size_t)nxt * (16 * D_MODEL));
            #pragma unroll
            for (int i = 0; i < 24; ++i)
                __builtin_prefetch(p + (size_t)(i * 32 + lane) * 64);
        }

        // ---- LayerNorm statistics (coalesced b128 loads + wave32 butterfly) ----
        #pragma unroll 1
        for (int t = 0; t < 16; ++t) {
            const float* row = xt + t * D_MODEL;
            float s = 0.f, ss = 0.f;
            #pragma unroll
            for (int j = 0; j < 6; ++j) {           // 6 * 32 lanes * 4 floats = 768
                v4f v = *(const v4f*)(row + j * 128 + lane * 4);
                s  += (v.x + v.y) + (v.z + v.w);
                ss += (v.x * v.x + v.y * v.y) + (v.z * v.z + v.w * v.w);
            }
            #pragma unroll
            for (int off = 16; off > 0; off >>= 1) {
                s  += __shfl_xor(s,  off, 32);
                ss += __shfl_xor(ss, off, 32);
            }
            float mean = s * (1.0f / 768.0f);
            float var  = ss * (1.0f / 768.0f) - mean * mean;
            if (lane == 0) { ldsMu[t] = mean; ldsRs[t] = rsqrtf(var + LN_EPS); }
        }

        // ---- Down-projection: 192 x v_wmma_f32_16x16x4_f32 on raw x ----
        // A(16x4) layout: lane -> token = lane&15, holds K = kb*4 + 2*halfSel + {0,1}
        v8f acc0 = {0.f,0.f,0.f,0.f,0.f,0.f,0.f,0.f};
        v8f acc1 = acc0, acc2 = acc0, acc3 = acc0;
        const float* aBase = xt + (size_t)laneN * D_MODEL + halfSel * 2;
        const float* bBase = smem + OFF_WG + lane * 2;
        #pragma unroll 4
        for (int kb = 0; kb < 192; kb += 4) {
            v2f a0 = *(const v2f*)(aBase + (kb + 0) * 4);
            v2f b0 = *(const v2f*)(bBase + (kb + 0) * 64);
            acc0 = __builtin_amdgcn_wmma_f32_16x16x4_f32(false, a0, false, b0, (short)0, acc0, false, false);
            v2f a1 = *(const v2f*)(aBase + (kb + 1) * 4);
            v2f b1 = *(const v2f*)(bBase + (kb + 1) * 64);
            acc1 = __builtin_amdgcn_wmma_f32_16x16x4_f32(false, a1, false, b1, (short)0, acc1, false, false);
            v2f a2 = *(const v2f*)(aBase + (kb + 2) * 4);
            v2f b2 = *(const v2f*)(bBase + (kb + 2) * 64);
            acc2 = __builtin_amdgcn_wmma_f32_16x16x4_f32(false, a2, false, b2, (short)0, acc2, false, false);
            v2f a3 = *(const v2f*)(aBase + (kb + 3) * 4);
            v2f b3 = *(const v2f*)(bBase + (kb + 3) * 64);
            acc3 = __builtin_amdgcn_wmma_f32_16x16x4_f32(false, a3, false, b3, (short)0, acc3, false, false);
        }
        v8f accd = (acc0 + acc1) + (acc2 + acc3);

        // Epilogue: down = relu(rstd*acc - rstd*mu*c1 + c2); park 16x16 in LDS.
        // C/D layout: VGPR j holds (M = j + 8*halfSel, N = lane&15).
        #pragma unroll
        for (int j = 0; j < 8; ++j) {
            int m = j + 8 * halfSel;
            float mu = ldsMu[m], rs = ldsRs[m];
            float v = rs * accd[j] - rs * mu * c1k + c2k;
            ldsDown[m * 16 + laneN] = fmaxf(v, 0.f);
        }

        // ---- Up-projection: 48 N-tiles x 4 WMMAs (K=16), then scale+residual ----
        const float* aUp = ldsDown + laneN * 16 + halfSel * 2;
        const float* bUp = smem + OFF_WU + lane * 2;
        #pragma unroll 1
        for (int nt = 0; nt < 48; ++nt) {
            v8f u = {0.f,0.f,0.f,0.f,0.f,0.f,0.f,0.f};
            #pragma unroll
            for (int kb = 0; kb < 4; ++kb) {
                v2f a = *(const v2f*)(aUp + kb * 4);
                v2f b = *(const v2f*)(bUp + (nt * 4 + kb) * 64);
                u = __builtin_amdgcn_wmma_f32_16x16x4_f32(false, a, false, b, (short)0, u, false, false);
            }
            int n = nt * 16 + laneN;
            float bu = smem[OFF_BUP + n];
            #pragma unroll
            for (int j = 0; j < 8; ++j) {
                int m = j + 8 * halfSel;
                size_t g = (size_t)(tile * 16 + m) * D_MODEL + n;
                out[g] = x[g] + SCALE_F * (u[j] + bu);   // x re-read hits WGP$/L2
            }
        }
    }
}

extern "C" void kernel_launch(void* const* d_in, const int* in_sizes, int n_in,
                              void* d_out, int out_size, void* d_ws, size_t ws_size,
                              hipStream_t stream) {
    (void)n_in; (void)out_size; (void)d_ws; (void)ws_size;
    const float* x     = (const float*)d_in[0];
    const float* gamma = (const float*)d_in[1];
    const float* beta  = (const float*)d_in[2];
    const float* wdown = (const float*)d_in[3];
    const float* bdown = (const float*)d_in[4];
    const float* wup   = (const float*)d_in[5];
    const float* bup   = (const float*)d_in[6];
    float* out = (float*)d_out;

    const int nTokens = in_sizes[0] / D_MODEL;      // 8*4096 = 32768
    dim3 grid(128), block(256);                     // 1024 waves -> 2 tiles/wave
    size_t shmem = (size_t)LDS_FLOATS * sizeof(float);  // 110720 B (<320KB WGP LDS)
    hipLaunchKernelGGL(adapter_wmma_kernel, grid, block, shmem, stream,
                       x, gamma, beta, wdown, bdown, wup, bup, out, nTokens);
}